// char_func_path_3100966388053
// MI455X (gfx1250) — compile-verified
//
#include <hip/hip_runtime.h>

// Problem constants (from reference setup_inputs)
#define NPATH 64
#define TLEN  32
#define DDIM  8
#define CCH   16
#define MDIM  32
#define MSQ   (MDIM*MDIM)   // 1024 elements per 32x32 plane

typedef __attribute__((ext_vector_type(2))) float v2f;
typedef __attribute__((ext_vector_type(8))) float v8f;

// CDNA5 f32 matrix op: D(16x16) = A(16x4) * B(4x16) + C, wave32.
__device__ __forceinline__ v8f wmma4(v2f a, v2f b, v8f c) {
  return __builtin_amdgcn_wmma_f32_16x16x4_f32(false, a, false, b, (short)0, c,
                                               false, false);
}

// One wave computes D = A * B for 32x32 complex matrices resident in LDS.
// Layouts per ISA 7.12.2 (f32):
//   A frag (16x4): lane<16 -> {A[r][k0],A[r][k0+1]}, lane>=16 -> {A[r][k0+2],A[r][k0+3]}
//   B frag (4x16): mirrored K split across lane halves, N = lane&15
//   C/D (16x16):  vgpr v -> row v (lanes 0-15) / row v+8 (lanes 16-31), col = lane&15
__device__ __forceinline__ void cmatmul32(const float* Are, const float* Aim,
                                          const float* Bre, const float* Bim,
                                          float* Dre, float* Dim, int lane) {
  const int half = lane >> 4;
  const int l15  = lane & 15;
#pragma unroll
  for (int ti = 0; ti < 2; ++ti) {
#pragma unroll
    for (int tj = 0; tj < 2; ++tj) {
      v8f accr = {0.f,0.f,0.f,0.f,0.f,0.f,0.f,0.f};
      v8f acci = {0.f,0.f,0.f,0.f,0.f,0.f,0.f,0.f};
#pragma unroll
      for (int kb = 0; kb < 8; ++kb) {
        const int ks   = kb*4 + half*2;
        const int arow = ti*16 + l15;
        const int bcol = tj*16 + l15;
        v2f ar, ai, br, bi;
        ar[0] = Are[arow*MDIM + ks];     ar[1] = Are[arow*MDIM + ks + 1];
        ai[0] = Aim[arow*MDIM + ks];     ai[1] = Aim[arow*MDIM + ks + 1];
        br[0] = Bre[ks*MDIM + bcol];     br[1] = Bre[(ks+1)*MDIM + bcol];
        bi[0] = Bim[ks*MDIM + bcol];     bi[1] = Bim[(ks+1)*MDIM + bcol];
        // (Ar + iAi)(Br + iBi): re = ArBr - AiBi ; im = ArBi + AiBr
        accr = wmma4(ar,  br, accr);
        accr = wmma4(-ai, bi, accr);   // minus via VALU negate (f32 WMMA NEG is C-only)
        acci = wmma4(ar,  bi, acci);
        acci = wmma4(ai,  br, acci);
      }
      const int rbase = ti*16 + half*8;
      const int cidx  = tj*16 + l15;
#pragma unroll
      for (int r = 0; r < 8; ++r) {
        Dre[(rbase + r)*MDIM + cidx] = accr[r];
        Dim[(rbase + r)*MDIM + cidx] = acci[r];
      }
    }
  }
}

// Block = 4 waves sharing channel c (staged A_anti[c] in LDS), each wave owns one path n.
// Dynamic LDS: A_anti (2*8*1024 f32 = 64KB) + 4 waves * 8 planes * 4KB = 128KB -> 192KB.
__global__ void __launch_bounds__(128, 1)
develop_kernel(const float* __restrict__ X1, const float* __restrict__ X2,
               const float* __restrict__ Ar, const float* __restrict__ Ai,
               float* __restrict__ Dm_re, float* __restrict__ Dm_im,
               float* __restrict__ Di_re, float* __restrict__ Di_im) {
  extern __shared__ float smem[];
  float* Aa_re = smem;                 // [DDIM][MSQ]
  float* Aa_im = Aa_re + DDIM*MSQ;

  const int tid  = threadIdx.x;
  const int wave = tid >> 5;
  const int lane = tid & 31;
  const int c    = blockIdx.y;
  const int set  = blockIdx.z;
  const int n    = blockIdx.x*4 + wave;

  // Stage anti-Hermitian generators for this channel:
  //   re: Ar[i][j] - Ar[j][i] ; im: Ai[i][j] + Ai[j][i]
  const int abase = c*DDIM*MSQ;
  for (int e = tid; e < DDIM*MSQ; e += 128) {
    const int dd = e >> 10, rem = e & (MSQ-1);
    const int i = rem >> 5, j = rem & 31;
    const float arij = Ar[abase + dd*MSQ + i*MDIM + j];
    const float arji = Ar[abase + dd*MSQ + j*MDIM + i];
    const float aiij = Ai[abase + dd*MSQ + i*MDIM + j];
    const float aiji = Ai[abase + dd*MSQ + j*MDIM + i];
    Aa_re[e] = arij - arji;
    Aa_im[e] = aiij + aiji;
  }
  __syncthreads();

  float* wbase = Aa_im + DDIM*MSQ + wave*8*MSQ;
  float* S_re = wbase;         float* S_im = wbase + 1*MSQ;
  float* Q_re = wbase + 2*MSQ; float* Q_im = wbase + 3*MSQ;
  float* P_re = wbase + 4*MSQ; float* P_im = wbase + 5*MSQ;
  float* U_re = wbase + 6*MSQ; float* U_im = wbase + 7*MSQ;

  const float* Xn = (set ? X2 : X1) + n*TLEN*DDIM;

  // U = I
  for (int i = 0; i < MDIM; ++i) {
    const int idx = i*MDIM + lane;
    U_re[idx] = (i == lane) ? 1.f : 0.f;
    U_im[idx] = 0.f;
  }

  const float w = (set == 0 ? 1.f : -1.f) * (1.f/(float)NPATH);

  for (int t = 0; t < TLEN; ++t) {
    // path increment (include_initial: dX_0 = X[:,0])
    float dx[DDIM];
#pragma unroll
    for (int dd = 0; dd < DDIM; ++dd) {
      const float cur  = Xn[t*DDIM + dd];
      const float prev = (t == 0) ? 0.f : Xn[(t-1)*DDIM + dd];
      dx[dd] = cur - prev;
    }
    // S = (1/2^4) * sum_d dx[d] * A_anti[d]   (pre-scaled for squaring)
    for (int i = 0; i < MDIM; ++i) {
      const int idx = i*MDIM + lane;
      float sr = 0.f, si = 0.f;
#pragma unroll
      for (int dd = 0; dd < DDIM; ++dd) {
        sr += dx[dd]*Aa_re[dd*MSQ + idx];
        si += dx[dd]*Aa_im[dd*MSQ + idx];
      }
      S_re[idx] = sr * 0.0625f;
      S_im[idx] = si * 0.0625f;
    }
    // expm(16*S) via Taylor-6 Horner on S, then 4 squarings.
    // Q = I + S/6
    for (int i = 0; i < MDIM; ++i) {
      const int idx = i*MDIM + lane;
      const float dg = (i == lane) ? 1.f : 0.f;
      Q_re[idx] = S_re[idx]*(1.f/6.f) + dg;
      Q_im[idx] = S_im[idx]*(1.f/6.f);
    }
#pragma unroll 1
    for (int k = 5; k >= 1; --k) {           // Q = I + (S*Q)/k
      cmatmul32(S_re, S_im, Q_re, Q_im, P_re, P_im, lane);
      const float invk = 1.f/(float)k;
      for (int i = 0; i < MDIM; ++i) {
        const int idx = i*MDIM + lane;
        const float dg = (i == lane) ? 1.f : 0.f;
        Q_re[idx] = P_re[idx]*invk + dg;
        Q_im[idx] = P_im[idx]*invk;
      }
    }
#pragma unroll 1
    for (int s = 0; s < 2; ++s) {            // Q <- Q^4, twice => Q^(16)
      cmatmul32(Q_re, Q_im, Q_re, Q_im, P_re, P_im, lane);
      cmatmul32(P_re, P_im, P_re, P_im, Q_re, Q_im, lane);
    }
    // initial-increment term: develop([0, X[:,0]]) == E_{t=0}
    if (t == 0) {
      for (int i = 0; i < MDIM; ++i) {
        const int idx = i*MDIM + lane;
        atomicAdd(&Di_re[c*MSQ + idx], w*Q_re[idx]);
        atomicAdd(&Di_im[c*MSQ + idx], w*Q_im[idx]);
      }
    }
    // U = U @ E
    cmatmul32(U_re, U_im, Q_re, Q_im, P_re, P_im, lane);
    for (int i = 0; i < MDIM; ++i) {
      const int idx = i*MDIM + lane;
      U_re[idx] = P_re[idx];
      U_im[idx] = P_im[idx];
    }
  }

  // CF accumulation: D += +/- (1/N) * U   (CF1 - CF2 fused by sign)
  for (int i = 0; i < MDIM; ++i) {
    const int idx = i*MDIM + lane;
    atomicAdd(&Dm_re[c*MSQ + idx], w*U_re[idx]);
    atomicAdd(&Dm_im[c*MSQ + idx], w*U_im[idx]);
  }
}

__global__ void zero_kernel(float* p, int n) {
  const int i = blockIdx.x*blockDim.x + threadIdx.x;
  if (i < n) p[i] = 0.f;
}

// out = mean_c ||Dm[c]||_F^2 + 0.1 * mean_c ||Di[c]||_F^2
__global__ void reduce_kernel(const float* __restrict__ ws, float* __restrict__ out) {
  __shared__ float red[256];
  float am = 0.f, ai = 0.f;
  for (int i = threadIdx.x; i < 2*CCH*MSQ; i += 256) { const float v = ws[i];             am += v*v; }
  for (int i = threadIdx.x; i < 2*CCH*MSQ; i += 256) { const float v = ws[2*CCH*MSQ + i]; ai += v*v; }
  red[threadIdx.x] = am*(1.f/(float)CCH) + 0.1f*ai*(1.f/(float)CCH);
  __syncthreads();
  for (int s = 128; s > 0; s >>= 1) {
    if (threadIdx.x < s) red[threadIdx.x] += red[threadIdx.x + s];
    __syncthreads();
  }
  if (threadIdx.x == 0) out[0] = red[0];
}

extern "C" void kernel_launch(void* const* d_in, const int* in_sizes, int n_in,
                              void* d_out, int out_size, void* d_ws, size_t ws_size,
                              hipStream_t stream) {
  (void)in_sizes; (void)n_in; (void)out_size; (void)ws_size;
  const float* X1 = (const float*)d_in[0];
  const float* X2 = (const float*)d_in[1];
  const float* Ar = (const float*)d_in[2];
  const float* Ai = (const float*)d_in[3];

  float* ws    = (float*)d_ws;
  float* Dm_re = ws;
  float* Dm_im = ws + 1*CCH*MSQ;
  float* Di_re = ws + 2*CCH*MSQ;
  float* Di_im = ws + 3*CCH*MSQ;

  const int nz = 4*CCH*MSQ;   // 65536 floats = 256 KB of accumulators
  zero_kernel<<<(nz + 255)/256, 256, 0, stream>>>(ws, nz);

  const size_t shmem = (size_t)(2*DDIM*MSQ + 4*8*MSQ) * sizeof(float); // 192 KB < 320 KB/WGP
  hipFuncSetAttribute(reinterpret_cast<const void*>(develop_kernel),
                      hipFuncAttributeMaxDynamicSharedMemorySize, (int)shmem);

  dim3 grid(NPATH/4, CCH, 2);   // 16 x 16 x 2 = 512 blocks of 4 waves
  develop_kernel<<<grid, 128, shmem, stream>>>(X1, X2, Ar, Ai,
                                               Dm_re, Dm_im, Di_re, Di_im);

  reduce_kernel<<<1, 256, 0, stream>>>(ws, (float*)d_out);
}